// Net_54288386621899
// MI455X (gfx1250) — compile-verified
//
#include <hip/hip_runtime.h>

#define BATCH 32
#define PTS   2048
#define NK    32
#define NB    33   // K neighbors + self loop

typedef float v2f __attribute__((ext_vector_type(2)));
typedef float v8f __attribute__((ext_vector_type(8)));

// ---------------------------------------------------------------------------
// Radius top-K neighbor search. One thread per query point; register-resident
// sorted (ascending d2) top-32 within radius, self excluded; slot 32 = self.
// Invalid slots = -1.
// ---------------------------------------------------------------------------
__global__ __launch_bounds__(256)
void neighbors_kernel(const float* __restrict__ pos, int n, int total, float r2,
                      int* __restrict__ idx_out) {
    int gid = blockIdx.x * blockDim.x + threadIdx.x;
    if (gid >= total) return;
    int c = gid / n, i = gid - c * n;
    const float* pc = pos + (size_t)c * n * 3;
    float xi = pc[i * 3 + 0], yi = pc[i * 3 + 1], zi = pc[i * 3 + 2];

    float nd[NK]; int ni[NK];
#pragma unroll
    for (int q = 0; q < NK; ++q) { nd[q] = __builtin_inff(); ni[q] = -1; }

    for (int j = 0; j < n; ++j) {
        float dx = pc[j * 3 + 0] - xi;
        float dy = pc[j * 3 + 1] - yi;
        float dz = pc[j * 3 + 2] - zi;
        float d2 = dx * dx + dy * dy + dz * dz;
        if (j == i || d2 > r2) continue;
        if (d2 < nd[NK - 1]) {
            nd[NK - 1] = d2; ni[NK - 1] = j;
            // single bubble pass restores sorted order
#pragma unroll
            for (int q = NK - 1; q > 0; --q) {
                if (nd[q] < nd[q - 1]) {
                    float tf = nd[q]; nd[q] = nd[q - 1]; nd[q - 1] = tf;
                    int   ti = ni[q]; ni[q] = ni[q - 1]; ni[q - 1] = ti;
                }
            }
        }
    }
    int* o = idx_out + (size_t)gid * NB;
#pragma unroll
    for (int q = 0; q < NK; ++q) o[q] = ni[q];
    o[NK] = i;  // self loop, always valid
}

// ---------------------------------------------------------------------------
// Farthest point sampling: one block per cloud, coords + min-dist in LDS.
// Argmax tie-break = lowest index (matches jnp.argmax).
// ---------------------------------------------------------------------------
__global__ __launch_bounds__(256)
void fps_kernel(const float* __restrict__ pos, int n_in, int n_out,
                int* __restrict__ out) {
    __shared__ float sx[PTS], sy[PTS], sz[PTS], sd[PTS];
    __shared__ float rbv[8];
    __shared__ int   rbi[8];
    __shared__ int   s_cur;
    int c = blockIdx.x, tid = threadIdx.x;
    const float* pc = pos + (size_t)c * n_in * 3;

    for (int j = tid; j < n_in; j += 256) {
        sx[j] = pc[j * 3 + 0]; sy[j] = pc[j * 3 + 1]; sz[j] = pc[j * 3 + 2];
    }
    __syncthreads();
    float x0 = sx[0], y0 = sy[0], z0 = sz[0];
    for (int j = tid; j < n_in; j += 256) {
        float dx = sx[j] - x0, dy = sy[j] - y0, dz = sz[j] - z0;
        sd[j] = dx * dx + dy * dy + dz * dz;
    }
    int* oc = out + (size_t)c * n_out;
    if (tid == 0) oc[0] = 0;
    __syncthreads();

    for (int t = 1; t < n_out; ++t) {
        float bv = -1.0f; int bi = 0x7fffffff;
        for (int j = tid; j < n_in; j += 256) {
            float v = sd[j];
            if (v > bv || (v == bv && j < bi)) { bv = v; bi = j; }
        }
        for (int off = 16; off > 0; off >>= 1) {
            float ov = __shfl_down(bv, off, 32);
            int   oi = __shfl_down(bi, off, 32);
            if (ov > bv || (ov == bv && oi < bi)) { bv = ov; bi = oi; }
        }
        if ((tid & 31) == 0) { rbv[tid >> 5] = bv; rbi[tid >> 5] = bi; }
        __syncthreads();
        if (tid == 0) {
            float v = rbv[0]; int b = rbi[0];
            for (int w = 1; w < 8; ++w)
                if (rbv[w] > v || (rbv[w] == v && rbi[w] < b)) { v = rbv[w]; b = rbi[w]; }
            oc[t] = b; s_cur = b;
        }
        __syncthreads();
        int cur = s_cur;
        float cx = sx[cur], cy = sy[cur], cz = sz[cur];
        for (int j = tid; j < n_in; j += 256) {
            float dx = sx[j] - cx, dy = sy[j] - cy, dz = sz[j] - cz;
            float d2 = dx * dx + dy * dy + dz * dz;
            if (d2 < sd[j]) sd[j] = d2;
        }
        __syncthreads();
    }
}

// ---------------------------------------------------------------------------
// Compact pos/x through FPS sample indices.
// ---------------------------------------------------------------------------
__global__ __launch_bounds__(256)
void gather_kernel(const float* __restrict__ pos_in, const float* __restrict__ x_in,
                   const int* __restrict__ samp, float* __restrict__ pos_out,
                   float* __restrict__ x_out, int n_in, int n_out, int F) {
    int gid = blockIdx.x * blockDim.x + threadIdx.x;
    if (gid >= BATCH * n_out) return;
    int c = gid / n_out, i = gid - c * n_out;
    int s = samp[(size_t)c * n_out + i];
    const float* ps = pos_in + ((size_t)c * n_in + s) * 3;
    float* pd = pos_out + ((size_t)c * n_out + i) * 3;
    pd[0] = ps[0]; pd[1] = ps[1]; pd[2] = ps[2];
    const float* xs = x_in + ((size_t)c * n_in + s) * F;
    float* xd = x_out + ((size_t)c * n_out + i) * F;
    for (int f = 0; f < F; ++f) xd[f] = xs[f];
}

// ---------------------------------------------------------------------------
// Fused PointConv: one workgroup (128 threads / 4 waves) per point.
//   feats[48][CINP] (rows = 33 neighbors, zero-padded) -> WMMA f32 GEMM1
//   H = relu(F@W1+b1) in LDS -> WMMA f32 GEMM2 -> masked row-max -> relu.
// K loop is outermost per n-tile: the B fragment (shared by all 3 M-tiles) is
// loaded once per K-step and fed to 3 back-to-back WMMAs into 3 accumulators.
// Only the final K-tile can touch padded W1 rows; that iteration is peeled and
// handled with branch-free clamped loads + selects (no exec-mask branches).
// A-frag:  lane<16: M=lane, v0=K0,v1=K1 ; lane>=16: M=lane-16, v0=K2,v1=K3
// B-frag:  lane<16: N=lane, v0=K0,v1=K1 ; lane>=16: N=lane-16, v0=K2,v1=K3
// C/D:     vgpr v: lanes<16 -> (M=v, N=lane); lanes>=16 -> (M=v+8, N=lane-16)
// ---------------------------------------------------------------------------
template <int CIN, int CINP, int CHID, int COUT>
__global__ __launch_bounds__(128)
void point_conv_kernel(const float* __restrict__ pos, const float* __restrict__ xin,
                       const int* __restrict__ idx,
                       const float* __restrict__ W1, const float* __restrict__ b1,
                       const float* __restrict__ W2, const float* __restrict__ b2,
                       float* __restrict__ xout, int n) {
    constexpr int ROWS  = 48;          // 33 rows padded to 3 M-tiles
    constexpr int NPREV = CIN - 3;
    constexpr int KT1   = CINP / 4;
    constexpr int NT1   = CHID / 16;
    constexpr int KT2   = CHID / 4;
    constexpr int NT2   = COUT / 16;

    extern __shared__ float smem[];
    float* sF = smem;                        // ROWS * CINP
    float* sH = smem + ROWS * CINP;          // ROWS * CHID
    int*   sV = (int*)(sH + ROWS * CHID);    // ROWS validity flags

    const int tid  = threadIdx.x;
    const int wave = tid >> 5;
    const int lane = tid & 31;
    const int l15  = lane & 15;
    const bool hi  = lane >= 16;
    const int koff = hi ? 2 : 0;

    const int c = blockIdx.x / n;
    const int i = blockIdx.x - c * n;
    const float* posC = pos + (size_t)c * n * 3;
    const int*   nbr  = idx + ((size_t)c * n + i) * NB;

    const float cx = posC[i * 3 + 0];
    const float cy = posC[i * 3 + 1];
    const float cz = posC[i * 3 + 2];

    for (int e = tid; e < ROWS * CINP; e += 128) sF[e] = 0.0f;
    for (int r = tid; r < ROWS; r += 128) sV[r] = (r < NB && nbr[r] >= 0) ? 1 : 0;
    __syncthreads();

    // gather feats = concat(x_j, pos_j - pos_i)
    for (int e = tid; e < NB * CIN; e += 128) {
        int r = e / CIN, f = e - r * CIN;
        int j = nbr[r];
        if (j >= 0) {
            float v;
            if (f < NPREV) {
                v = xin[((size_t)c * n + j) * NPREV + f];
            } else {
                int d = f - NPREV;
                v = posC[j * 3 + d] - (d == 0 ? cx : (d == 1 ? cy : cz));
            }
            sF[r * CINP + f] = v;
        }
    }
    __syncthreads();

    const int r0 = 0 * 16 + l15;
    const int r1 = 1 * 16 + l15;
    const int r2 = 2 * 16 + l15;

    // ---- GEMM1: H = relu(F @ W1 + b1) --------------------------------------
    for (int nt = wave; nt < NT1; nt += 4) {        // wave-uniform N split
        const int col = nt * 16 + l15;
        const float bias = b1[col];
        v8f acc0 = {}, acc1 = {}, acc2 = {};
        for (int kt = 0; kt < KT1 - 1; ++kt) {      // fully in-bounds K-tiles
            const int kb = kt * 4 + koff;
            v2f b, a0, a1, a2;
            b[0] = W1[(size_t)kb * CHID + col];
            b[1] = W1[(size_t)(kb + 1) * CHID + col];
            a0[0] = sF[r0 * CINP + kb]; a0[1] = sF[r0 * CINP + kb + 1];
            a1[0] = sF[r1 * CINP + kb]; a1[1] = sF[r1 * CINP + kb + 1];
            a2[0] = sF[r2 * CINP + kb]; a2[1] = sF[r2 * CINP + kb + 1];
            acc0 = __builtin_amdgcn_wmma_f32_16x16x4_f32(false, a0, false, b,
                                                         (short)0, acc0, false, false);
            acc1 = __builtin_amdgcn_wmma_f32_16x16x4_f32(false, a1, false, b,
                                                         (short)0, acc1, false, false);
            acc2 = __builtin_amdgcn_wmma_f32_16x16x4_f32(false, a2, false, b,
                                                         (short)0, acc2, false, false);
        }
        {   // peeled last K-tile: branch-free handling of padded W1 rows
            const int kb = (KT1 - 1) * 4 + koff;
            const int k0 = (kb     < CIN) ? kb     : (CIN - 1);  // clamped, in-bounds
            const int k1 = (kb + 1 < CIN) ? kb + 1 : (CIN - 1);
            float w0 = W1[(size_t)k0 * CHID + col];
            float w1 = W1[(size_t)k1 * CHID + col];
            v2f b, a0, a1, a2;
            b[0] = (kb     < CIN) ? w0 : 0.0f;
            b[1] = (kb + 1 < CIN) ? w1 : 0.0f;
            a0[0] = sF[r0 * CINP + kb]; a0[1] = sF[r0 * CINP + kb + 1];
            a1[0] = sF[r1 * CINP + kb]; a1[1] = sF[r1 * CINP + kb + 1];
            a2[0] = sF[r2 * CINP + kb]; a2[1] = sF[r2 * CINP + kb + 1];
            acc0 = __builtin_amdgcn_wmma_f32_16x16x4_f32(false, a0, false, b,
                                                         (short)0, acc0, false, false);
            acc1 = __builtin_amdgcn_wmma_f32_16x16x4_f32(false, a1, false, b,
                                                         (short)0, acc1, false, false);
            acc2 = __builtin_amdgcn_wmma_f32_16x16x4_f32(false, a2, false, b,
                                                         (short)0, acc2, false, false);
        }
#pragma unroll
        for (int mt = 0; mt < 3; ++mt) {
            v8f acc = (mt == 0) ? acc0 : ((mt == 1) ? acc1 : acc2);
            const int rbase = mt * 16 + (hi ? 8 : 0);
#pragma unroll
            for (int v = 0; v < 8; ++v) {
                float h = acc[v] + bias;
                sH[(rbase + v) * CHID + col] = h > 0.0f ? h : 0.0f;
            }
        }
    }
    __syncthreads();

    // ---- GEMM2: O = H @ W2 + b2, masked max over rows, outer relu ---------
    float* xo = xout + ((size_t)c * n + i) * COUT;
    for (int nt = wave; nt < NT2; nt += 4) {
        const int col = nt * 16 + l15;
        const float bias = b2[col];
        v8f acc0 = {}, acc1 = {}, acc2 = {};
        for (int kt = 0; kt < KT2; ++kt) {          // CHID % 4 == 0: no guards
            const int kb = kt * 4 + koff;
            v2f b, a0, a1, a2;
            b[0] = W2[(size_t)kb * COUT + col];
            b[1] = W2[(size_t)(kb + 1) * COUT + col];
            a0[0] = sH[r0 * CHID + kb]; a0[1] = sH[r0 * CHID + kb + 1];
            a1[0] = sH[r1 * CHID + kb]; a1[1] = sH[r1 * CHID + kb + 1];
            a2[0] = sH[r2 * CHID + kb]; a2[1] = sH[r2 * CHID + kb + 1];
            acc0 = __builtin_amdgcn_wmma_f32_16x16x4_f32(false, a0, false, b,
                                                         (short)0, acc0, false, false);
            acc1 = __builtin_amdgcn_wmma_f32_16x16x4_f32(false, a1, false, b,
                                                         (short)0, acc1, false, false);
            acc2 = __builtin_amdgcn_wmma_f32_16x16x4_f32(false, a2, false, b,
                                                         (short)0, acc2, false, false);
        }
        float cmax = -__builtin_inff();
#pragma unroll
        for (int mt = 0; mt < 3; ++mt) {
            v8f acc = (mt == 0) ? acc0 : ((mt == 1) ? acc1 : acc2);
            const int rbase = mt * 16 + (hi ? 8 : 0);
#pragma unroll
            for (int v = 0; v < 8; ++v) {
                float h = acc[v] + bias;
                cmax = fmaxf(cmax, sV[rbase + v] ? h : -__builtin_inff());
            }
        }
        cmax = fmaxf(cmax, __shfl_xor(cmax, 16, 32));  // combine the two row halves
        if (!hi) xo[col] = cmax > 0.0f ? cmax : 0.0f;
    }
}

// ---------------------------------------------------------------------------
// Global max pool over the 256 surviving points (256 features each).
// ---------------------------------------------------------------------------
__global__ __launch_bounds__(256)
void gmax_kernel(const float* __restrict__ x, float* __restrict__ g) {
    int c = blockIdx.x, f = threadIdx.x;
    const float* xc = x + (size_t)c * 256 * 256;
    float m = -__builtin_inff();
    for (int p = 0; p < 256; ++p) m = fmaxf(m, xc[p * 256 + f]);
    g[c * 256 + f] = m;
}

// ---------------------------------------------------------------------------
// Classifier head: 256->256 relu, 256->256 relu, 256->40, log_softmax.
// ---------------------------------------------------------------------------
__global__ __launch_bounds__(256)
void head_kernel(const float* __restrict__ g,
                 const float* __restrict__ Wl1, const float* __restrict__ bl1,
                 const float* __restrict__ Wl2, const float* __restrict__ bl2,
                 const float* __restrict__ Wl3, const float* __restrict__ bl3,
                 float* __restrict__ out) {
    __shared__ float a[256], h[256], lg[40];
    __shared__ float s_mx, s_ls;
    int c = blockIdx.x, t = threadIdx.x;
    a[t] = g[c * 256 + t];
    __syncthreads();
    float acc = bl1[t];
    for (int k = 0; k < 256; ++k) acc += a[k] * Wl1[k * 256 + t];
    h[t] = fmaxf(acc, 0.0f);
    __syncthreads();
    acc = bl2[t];
    for (int k = 0; k < 256; ++k) acc += h[k] * Wl2[k * 256 + t];
    __syncthreads();
    a[t] = fmaxf(acc, 0.0f);
    __syncthreads();
    if (t < 40) {
        float l = bl3[t];
        for (int k = 0; k < 256; ++k) l += a[k] * Wl3[k * 40 + t];
        lg[t] = l;
    }
    __syncthreads();
    if (t == 0) {
        float mx = lg[0];
        for (int q = 1; q < 40; ++q) mx = fmaxf(mx, lg[q]);
        float s = 0.0f;
        for (int q = 0; q < 40; ++q) s += expf(lg[q] - mx);
        s_mx = mx; s_ls = logf(s);
    }
    __syncthreads();
    if (t < 40) out[c * 40 + t] = lg[t] - s_mx - s_ls;
}

// ---------------------------------------------------------------------------
extern "C" void kernel_launch(void* const* d_in, const int* in_sizes, int n_in,
                              void* d_out, int out_size, void* d_ws, size_t ws_size,
                              hipStream_t stream) {
    (void)in_sizes; (void)n_in; (void)out_size; (void)ws_size;
    const float* pos = (const float*)d_in[0];
    // d_in[1] = batch (uniform repeat(arange(B),P)) -> unused
    const float* W1a = (const float*)d_in[2];  const float* b1a = (const float*)d_in[3];
    const float* W1b = (const float*)d_in[4];  const float* b1b = (const float*)d_in[5];
    const float* W2a = (const float*)d_in[6];  const float* b2a = (const float*)d_in[7];
    const float* W2b = (const float*)d_in[8];  const float* b2b = (const float*)d_in[9];
    const float* W3a = (const float*)d_in[10]; const float* b3a = (const float*)d_in[11];
    const float* W3b = (const float*)d_in[12]; const float* b3b = (const float*)d_in[13];
    const float* Wl1 = (const float*)d_in[14]; const float* bl1 = (const float*)d_in[15];
    const float* Wl2 = (const float*)d_in[16]; const float* bl2 = (const float*)d_in[17];
    const float* Wl3 = (const float*)d_in[18]; const float* bl3 = (const float*)d_in[19];
    float* out = (float*)d_out;
    char* ws = (char*)d_ws;

    // workspace layout (regions reused level-to-level; ~51 MB peak)
    size_t off = 0;
    auto carve = [&](size_t bytes) { size_t o = off; off += (bytes + 255) & ~(size_t)255; return o; };
    size_t o_idx  = carve((size_t)BATCH * PTS * NB * 4);     // neighbor idx, reused 3x
    size_t o_x1   = carve((size_t)BATCH * PTS * 64 * 4);     // x1; later reused for x3
    size_t o_xin  = carve((size_t)BATCH * 1024 * 64 * 4);    // gathered x input (L2/L3)
    size_t o_x2   = carve((size_t)BATCH * 1024 * 128 * 4);   // x2
    size_t o_f1   = carve((size_t)BATCH * 1024 * 4);         // fps level-1 indices
    size_t o_f2   = carve((size_t)BATCH * 256 * 4);          // fps level-2 indices
    size_t o_p2   = carve((size_t)BATCH * 1024 * 3 * 4);     // pos2
    size_t o_p3   = carve((size_t)BATCH * 256 * 3 * 4);      // pos3
    size_t o_g    = carve((size_t)BATCH * 256 * 4);          // pooled features

    int*   idxb = (int*)(ws + o_idx);
    float* x1   = (float*)(ws + o_x1);
    float* xinb = (float*)(ws + o_xin);
    float* x2   = (float*)(ws + o_x2);
    int*   f1   = (int*)(ws + o_f1);
    int*   f2   = (int*)(ws + o_f2);
    float* p2   = (float*)(ws + o_p2);
    float* p3   = (float*)(ws + o_p3);
    float* gbuf = (float*)(ws + o_g);
    float* x3   = x1;  // x1 dead after level-2 gather; reuse (8.4MB <= 16.8MB)

    // ---- Level 1: r=0.2, 2048 pts, feats=rel(3) -> 64 -> 64 ----------------
    neighbors_kernel<<<(BATCH * PTS + 255) / 256, 256, 0, stream>>>(
        pos, PTS, BATCH * PTS, 0.04f, idxb);
    size_t sh1 = (size_t)48 * (4 + 64) * 4 + 48 * 4;
    point_conv_kernel<3, 4, 64, 64><<<BATCH * PTS, 128, sh1, stream>>>(
        pos, nullptr, idxb, W1a, b1a, W1b, b1b, x1, PTS);
    fps_kernel<<<BATCH, 256, 0, stream>>>(pos, PTS, 1024, f1);
    gather_kernel<<<(BATCH * 1024 + 255) / 256, 256, 0, stream>>>(
        pos, x1, f1, p2, xinb, PTS, 1024, 64);

    // ---- Level 2: r=0.4, 1024 pts, feats=concat(x(64),rel) -> 128 -> 128 ---
    neighbors_kernel<<<(BATCH * 1024 + 255) / 256, 256, 0, stream>>>(
        p2, 1024, BATCH * 1024, 0.16f, idxb);
    size_t sh2 = (size_t)48 * (68 + 128) * 4 + 48 * 4;
    point_conv_kernel<67, 68, 128, 128><<<BATCH * 1024, 128, sh2, stream>>>(
        p2, xinb, idxb, W2a, b2a, W2b, b2b, x2, 1024);
    fps_kernel<<<BATCH, 256, 0, stream>>>(p2, 1024, 256, f2);
    gather_kernel<<<(BATCH * 256 + 255) / 256, 256, 0, stream>>>(
        p2, x2, f2, p3, xinb, 1024, 256, 128);

    // ---- Level 3: r=1.0, 256 pts, feats=concat(x(128),rel) -> 256 -> 256 ---
    neighbors_kernel<<<(BATCH * 256 + 255) / 256, 256, 0, stream>>>(
        p3, 256, BATCH * 256, 1.0f, idxb);
    size_t sh3 = (size_t)48 * (132 + 256) * 4 + 48 * 4;
    point_conv_kernel<131, 132, 256, 256><<<BATCH * 256, 128, sh3, stream>>>(
        p3, xinb, idxb, W3a, b3a, W3b, b3b, x3, 256);

    // ---- Global max pool + classifier head --------------------------------
    gmax_kernel<<<BATCH, 256, 0, stream>>>(x3, gbuf);
    head_kernel<<<BATCH, 256, 0, stream>>>(gbuf, Wl1, bl1, Wl2, bl2, Wl3, bl3, out);
}